// DirichletLoss_57853209477343
// MI455X (gfx1250) — compile-verified
//
#include <hip/hip_runtime.h>
#include <hip/hip_bf16.h>
#include <math.h>

// Problem constants (fixed by the reference)
#define GB 8            // graphs
#define GN 2048         // nodes per graph
#define GD 64           // feature dim
#define BN (GB*GN)      // total rows
#define KC 128          // K-chunk staged in LDS
#define PK (KC + 14)    // padded LDS row stride (floats): 14 mod 64 -> conflict-free b64 ld/st
#define WAVES 4         // waves per workgroup (wave32)

typedef __attribute__((ext_vector_type(2))) float v2f;
typedef __attribute__((ext_vector_type(8))) float v8f;

#if defined(__HIP_DEVICE_COMPILE__) && __has_builtin(__builtin_amdgcn_wmma_f32_16x16x4_f32)
#define HAS_WMMA_F32 1
#else
#define HAS_WMMA_F32 0
#endif

#if defined(__HIP_DEVICE_COMPILE__) && \
    __has_builtin(__builtin_amdgcn_global_load_async_to_lds_b32)
#define HAS_ASYNC 1
#else
#define HAS_ASYNC 0
#endif

#define AS1 __attribute__((address_space(1)))
#define AS3 __attribute__((address_space(3)))

// ---------------------------------------------------------------------------
// Kernel 1: row-wise L2 normalize h -> f  (torch F.normalize, eps=1e-12)
// ---------------------------------------------------------------------------
__global__ __launch_bounds__(256) void normalize_rows(const float* __restrict__ h,
                                                      float* __restrict__ f) {
  const int row = blockIdx.x * blockDim.x + threadIdx.x;   // grid sized exactly BN
  const float4* __restrict__ src = (const float4*)(h + (size_t)row * GD);
  float4* __restrict__ dst = (float4*)(f + (size_t)row * GD);
  float4 v[16];
  float ss = 0.f;
#pragma unroll
  for (int i = 0; i < 16; ++i) {
    v[i] = src[i];
    ss += v[i].x * v[i].x + v[i].y * v[i].y + v[i].z * v[i].z + v[i].w * v[i].w;
  }
  const float inv = 1.0f / fmaxf(sqrtf(ss), 1e-12f);
#pragma unroll
  for (int i = 0; i < 16; ++i) {
    float4 o;
    o.x = v[i].x * inv; o.y = v[i].y * inv; o.z = v[i].z * inv; o.w = v[i].w * inv;
    dst[i] = o;
  }
}

// ---------------------------------------------------------------------------
// Kernel 2: per 16-row strip of L: acc[16x64] = L_strip @ f via v_wmma_f32_16x16x4_f32.
// f chunk staged TRANSPOSED in LDS: fchT[d][k] with padded stride PK so each B operand
// is one aligned, bank-conflict-free ds_load_b64 at (const base + immediate offset).
// ---------------------------------------------------------------------------
__global__ __launch_bounds__(WAVES * 32) void dirichlet_wmma(
    const float* __restrict__ lap_s, const float* __restrict__ lap_t,
    const float* __restrict__ f_s, const float* __restrict__ f_t,
    float* __restrict__ partials) {
  __shared__ __align__(16) float fchT[GD * PK];   // 64*142*4 = 36352 B

  const int side  = blockIdx.z;
  const int graph = blockIdx.y;
  const int grp   = blockIdx.x;
  const int tid   = threadIdx.x;
  const int wave  = tid >> 5;
  const int lane  = tid & 31;
  const int lhalf = lane >> 4;
  const int l16   = lane & 15;

  const float* __restrict__ lap = side ? lap_t : lap_s;
  const float* __restrict__ fg  = (side ? f_t : f_s) + (size_t)graph * GN * GD;

  const int strip = grp * WAVES + wave;
  const int i0    = strip * 16;

  // A-tile: one global_load_b64 per lane = exact fp32 16x4 A layout
  // (VGPR0 = K0 lanes 0-15 / K2 lanes 16-31, VGPR1 = K1/K3).
  const float* __restrict__ arow =
      lap + ((size_t)graph * GN + (size_t)(i0 + l16)) * (size_t)GN + 2 * lhalf;

  // B base into transposed chunk: element (d, k) at fchT[d*PK + k].
  // Per-lane constant base; all k2/t offsets are DS immediates.
  const float* __restrict__ bbase = fchT + l16 * PK + 2 * lhalf;

  // Staging mapping: thread owns fixed d = tid&63, k advancing by stride.
  const int sd = tid & 63;
  const int sk0 = tid >> 6;            // 0 or 1

  v8f acc[4];
#pragma unroll
  for (int t = 0; t < 4; ++t)
#pragma unroll
    for (int r = 0; r < 8; ++r) acc[t][r] = 0.f;

  for (int kc = 0; kc < GN; kc += KC) {
    __syncthreads();  // previous chunk fully consumed
#if HAS_ASYNC
    {
      // Transposing copy, no VGPR round-trip: per-lane src (row-major f) and
      // dst (d-major LDS) addresses; b32 granularity. 64 ops/thread/chunk.
      const float* gsrc = fg + (size_t)kc * GD + sd;
#pragma unroll 8
      for (int i = 0; i < KC / 2; ++i) {
        const int k = sk0 + 2 * i;
        AS1 int* gaddr =
            (AS1 int*)(unsigned long long)(const void*)(gsrc + (size_t)k * GD);
        AS3 int* laddr =
            (AS3 int*)(unsigned int)(unsigned long long)(const void*)
                (fchT + sd * PK + k);
        __builtin_amdgcn_global_load_async_to_lds_b32(gaddr, laddr, 0, 0);
      }
      asm volatile("s_wait_asynccnt 0x0" ::: "memory");
    }
#else
    {
      // Sync transpose: coalesced b32 global reads, conflict-free ds_store_b64.
      const float* __restrict__ gsrc = fg + (size_t)kc * GD + sd;
#pragma unroll 4
      for (int i = 0; i < KC / 4; ++i) {
        const int k = 2 * sk0 + 4 * i;            // even k; (k,k+1) pair
        float2 p;
        p.x = gsrc[(size_t)k * GD];
        p.y = gsrc[(size_t)(k + 1) * GD];
        *(float2*)(fchT + sd * PK + k) = p;       // 8B aligned (PK even, k even)
      }
    }
#endif
    __syncthreads();

    if (kc + KC < GN)  // uniform; hint next f chunk toward L2 (global_prefetch_b8)
      __builtin_prefetch(fg + (size_t)(kc + KC) * GD + (size_t)tid * 16, 0, 1);

    const float* __restrict__ a = arow + kc;
#if HAS_WMMA_F32
#pragma unroll 4
    for (int k2 = 0; k2 < KC; k2 += 4) {
      v2f A = *(const v2f*)(a + k2);              // global_load_b64, 8B aligned
#pragma unroll
      for (int t = 0; t < 4; ++t) {
        // one ds_load_b64: {f[kc+k2+2*lhalf, 16t+l16], f[kc+k2+1+2*lhalf, 16t+l16]}
        v2f Bv = *(const v2f*)(bbase + t * 16 * PK + k2);
        acc[t] = __builtin_amdgcn_wmma_f32_16x16x4_f32(
            /*neg_a=*/false, A, /*neg_b=*/false, Bv,
            /*c_mod=*/(short)0, acc[t], /*reuse_a=*/false, /*reuse_b=*/false);
      }
    }
#else
    for (int k2 = 0; k2 < KC; ++k2) {             // VALU fallback (C-layout accumulators)
#pragma unroll
      for (int t = 0; t < 4; ++t) {
        const float fv = fchT[(t * 16 + l16) * PK + k2];
#pragma unroll
        for (int r = 0; r < 8; ++r) {
          acc[t][r] += lap[((size_t)graph * GN + (size_t)(i0 + r + 8 * lhalf)) * (size_t)GN +
                           (size_t)(kc + k2)] * fv;
        }
      }
    }
#endif
  }

  // Epilogue: dot f_strip (C layout) against acc, fixed-order wave reduce.
  float partial = 0.f;
#pragma unroll
  for (int t = 0; t < 4; ++t)
#pragma unroll
    for (int r = 0; r < 8; ++r) {
      const float fv = fg[(size_t)(i0 + r + 8 * lhalf) * GD + t * 16 + l16];
      partial += fv * acc[t][r];
    }
#pragma unroll
  for (int off = 16; off > 0; off >>= 1) partial += __shfl_xor(partial, off, 32);
  if (lane == 0)
    partials[((size_t)side * GB + graph) * (GN / 16) + strip] = partial;
}

// ---------------------------------------------------------------------------
// Kernel 3: deterministic final reduction.
// ---------------------------------------------------------------------------
__global__ void finalize_loss(const float* __restrict__ partials,
                              const int* __restrict__ has_s,
                              const int* __restrict__ has_t,
                              float* __restrict__ out) {
  __shared__ float sg[16];
  const int t = threadIdx.x;
  if (t < 16) {
    float s = 0.f;
    const float* p = partials + (size_t)t * (GN / 16);
    for (int i = 0; i < GN / 16; ++i) s += p[i];   // fixed order -> deterministic
    sg[t] = s;
  }
  __syncthreads();
  if (t == 0) {
    float total = 0.f, count = 0.f;
    for (int b = 0; b < GB; ++b) {
      const bool valid = (has_s[b] != 0) && (has_t[b] != 0);
      const float per = 0.5f * (sg[b] + sg[8 + b]) * (1.0f / (float)GD);  // mean over D
      if (valid) { total += per; count += 1.f; }
    }
    out[0] = (count > 0.f) ? (total / fmaxf(count, 1.f)) : 0.f;
  }
}

// ---------------------------------------------------------------------------
extern "C" void kernel_launch(void* const* d_in, const int* in_sizes, int n_in,
                              void* d_out, int out_size, void* d_ws, size_t ws_size,
                              hipStream_t stream) {
  const int*   has_s = (const int*)d_in[0];
  const float* lap_s = (const float*)d_in[1];
  const float* h_s   = (const float*)d_in[2];
  // d_in[3] = batch_s (unused: equal-sized graphs)
  const int*   has_t = (const int*)d_in[4];
  const float* lap_t = (const float*)d_in[5];
  const float* h_t   = (const float*)d_in[6];
  // d_in[7] = batch_t (unused)

  float* w     = (float*)d_ws;
  float* f_s   = w;                                  // BN*GD floats (4 MB)
  float* f_t   = w + (size_t)BN * GD;                // BN*GD floats (4 MB)
  float* parts = w + 2 * (size_t)BN * GD;            // 2*GB*(GN/16) = 2048 floats

  normalize_rows<<<BN / 256, 256, 0, stream>>>(h_s, f_s);
  normalize_rows<<<BN / 256, 256, 0, stream>>>(h_t, f_t);

  dim3 grid(GN / 16 / WAVES, GB, 2);
  dirichlet_wmma<<<grid, WAVES * 32, 0, stream>>>(lap_s, lap_t, f_s, f_t, parts);

  finalize_loss<<<1, 32, 0, stream>>>(parts, has_s, has_t, (float*)d_out);
}